// SoftDTW_79422535238096
// MI455X (gfx1250) — compile-verified
//
#include <hip/hip_runtime.h>
#include <hip/hip_bf16.h>

// Soft-DTW, fused: WMMA-computed 16x16 distance tiles staged through LDS,
// anti-diagonal DP with register-carried recurrences. One block per batch.

typedef float v2f __attribute__((ext_vector_type(2)));
typedef float v8f __attribute__((ext_vector_type(8)));

#define T_LEN 1024
#define D_DIM 8
#define NT    64            // 16x16 tiles per side (T_LEN/16)
#define BIGV  1e10f

__global__ __launch_bounds__(1024)
void SoftDTW_fused_kernel(const float* __restrict__ pred,
                          const float* __restrict__ target,
                          float* __restrict__ lossOut) {
    // LDS: 2 tile-diagonals of dist (2*64KB) + 2 R-diagonals (~8KB) + norms (8KB)
    __shared__ float distT[2][NT][16][16];
    __shared__ float Rd[2][T_LEN + 1];
    __shared__ float pn2[T_LEN];
    __shared__ float tn2[T_LEN];

    const int b = blockIdx.x;
    const int t = threadIdx.x;
    const float* __restrict__ P = pred   + (size_t)b * T_LEN * D_DIM;
    const float* __restrict__ G = target + (size_t)b * T_LEN * D_DIM;

    // --- squared norms of every pred/target row (t in 0..1023 == T rows) ---
    {
        float sp = 0.0f, st = 0.0f;
        #pragma unroll
        for (int d = 0; d < D_DIM; ++d) {
            float p = P[t * D_DIM + d]; sp += p * p;
            float g = G[t * D_DIM + d]; st += g * g;
        }
        pn2[t] = sp;
        tn2[t] = st;
    }
    // --- init R diagonal buffers: everything BIG; index 0 is never rewritten ---
    for (int i = t; i <= T_LEN; i += 1024) {
        Rd[0][i] = BIGV; Rd[1][i] = BIGV;
    }
    __syncthreads();

    // scalarize the wave index so the WMMA tile loop is a uniform SALU loop
    // (EXEC stays all-ones at every v_wmma, as the ISA requires)
    const int wave = __builtin_amdgcn_readfirstlane(t >> 5);   // 0..31
    const int lane = t & 31;
    const int m    = lane & 15;     // M (A) / N (B,C/D) index
    const int half = lane >> 4;     // K-half selector
    const int koff = half * 2;      // K offset {0,2} per f32 16x16x4 layout

    // register-carried DP state for row i = t+1:
    //   r0 = R[i-1][j-1] of the upcoming diagonal, rl = R[i][j-1]
    float r0_reg = (t == 0) ? 0.0f : BIGV;   // R[0][0] = 0 seeds thread 0
    float rl_reg = BIGV;

    for (int s = 0; s <= 127; ++s) {
        if (s <= 126) {
            // ---- compute dist tiles on tile-diagonal s via WMMA (f32, K=4 x2) ----
            const int lo = (s > NT - 1) ? (s - (NT - 1)) : 0;
            const int hi = (s < NT - 1) ? s : (NT - 1);
            for (int ti = lo + wave; ti <= hi; ti += 32) {   // uniform loop
                const int tj = s - ti;
                const float* Arow = P + (ti * 16 + m) * D_DIM;  // A: lane=M, vgpr+half=K
                const float* Brow = G + (tj * 16 + m) * D_DIM;  // B=target^T: lane=N
                v2f a0 = { Arow[koff],     Arow[koff + 1]     };  // K = 0..3
                v2f a1 = { Arow[4 + koff], Arow[4 + koff + 1] };  // K = 4..7
                v2f b0 = { Brow[koff],     Brow[koff + 1]     };
                v2f b1 = { Brow[4 + koff], Brow[4 + koff + 1] };
                v8f c = {0.f,0.f,0.f,0.f,0.f,0.f,0.f,0.f};
                c = __builtin_amdgcn_wmma_f32_16x16x4_f32(false, a0, false, b0,
                                                          (short)0, c, false, false);
                c = __builtin_amdgcn_wmma_f32_16x16x4_f32(false, a1, false, b1,
                                                          (short)0, c, false, false);
                // C/D layout: VGPR v -> M = v + 8*half, lane -> N = m
                #pragma unroll
                for (int v = 0; v < 8; ++v) {
                    const int row = v + 8 * half;
                    distT[s & 1][ti][row][m] =
                        pn2[ti * 16 + row] + tn2[tj * 16 + m] - 2.0f * c[v];
                }
            }
        }
        __syncthreads();

        // ---- DP over cell anti-diagonals unlocked by this tile-diagonal ----
        const int cLo = 16 * s;
        int cHi = 16 * s + 15;
        if (cHi > 2 * T_LEN - 2) cHi = 2 * T_LEN - 2;
        for (int cd = cLo; cd <= cHi; ++cd) {
            const int k = cd + 2;                   // R-diagonal index
            float*       cur  = Rd[k & 1];
            const float* prev = Rd[(k + 1) & 1];
            const int i = t + 1;
            const int j = k - i;
            const float r1 = prev[i - 1];           // R[i-1][j] (neighbor's last write)
            float val = BIGV;
            if (j >= 1 && j <= T_LEN) {
                const int di = i - 1, dj = j - 1;
                const float d  = distT[((di >> 4) + (dj >> 4)) & 1]
                                      [di >> 4][di & 15][dj & 15];
                float mn  = fminf(r0_reg, fminf(r1, rl_reg));
                float sum = __expf(mn - r0_reg) + __expf(mn - r1) + __expf(mn - rl_reg);
                val = d + mn - __logf(sum);         // gamma = 1
            }
            cur[i] = val;
            r0_reg = r1;                            // becomes R[i-1][j-1] next diagonal
            rl_reg = val;                           // becomes R[i][j-1] next diagonal
            __syncthreads();
        }
    }

    if (t == 0) lossOut[b] = Rd[(2 * T_LEN) & 1][T_LEN];   // R[T][T]
}

__global__ void SoftDTW_reduce_kernel(const float* __restrict__ lossArr,
                                      float* __restrict__ out, int B) {
    if (threadIdx.x == 0) {
        float s = 0.0f;
        for (int b = 0; b < B; ++b) s += lossArr[b];
        out[0] = s / (float)B;
    }
}

extern "C" void kernel_launch(void* const* d_in, const int* in_sizes, int n_in,
                              void* d_out, int out_size, void* d_ws, size_t ws_size,
                              hipStream_t stream) {
    const float* pred   = (const float*)d_in[0];
    const float* target = (const float*)d_in[1];
    const int B = in_sizes[0] / (T_LEN * D_DIM);   // 64 for the reference shapes

    float* lossArr = (float*)d_ws;                 // B floats of scratch

    SoftDTW_fused_kernel<<<B, 1024, 0, stream>>>(pred, target, lossArr);
    SoftDTW_reduce_kernel<<<1, 32, 0, stream>>>(lossArr, (float*)d_out, B);
}